// StarSolver_57226144252761
// MI455X (gfx1250) — compile-verified
//
#include <hip/hip_runtime.h>
#include <math.h>

// ---------------------------------------------------------------------------
// StarSolver on gfx1250: banded-Toeplitz 129-tap convolution via
// V_WMMA_F32_16X16X4_F32 (full f32), fused interp + residual RMS.
// Round 3: TDM (TENSOR_LOAD_TO_LDS) stages the strided raw/blaze columns into
// LDS asynchronously, overlapped with the star-interp gather pass.
// ---------------------------------------------------------------------------

typedef __attribute__((ext_vector_type(2))) float        v2f;
typedef __attribute__((ext_vector_type(8))) float        v8f;
typedef __attribute__((ext_vector_type(4))) unsigned int v4u;
typedef __attribute__((ext_vector_type(4))) int          v4i;
typedef __attribute__((ext_vector_type(8))) int          v8i;

#define NXM    65536
#define NXD    2048
#define NSPEC  128
#define NLSF   129
#define CHUNK  2048                 // conv outputs per workgroup
#define LBUF   (CHUNK + NLSF)       // 2177 staged padded-inner values
#define LSFPAD 160                  // 16 zeros | 129 taps | 15 zeros
#define W0     5000.0f
#define DWM    (100.0f / 65535.0f)  // master grid spacing (linspace)
#define DLR    (98.0f / 2047.0f)    // nominal detector spacing

// ---- Tensor Data Mover: async strided-column -> contiguous LDS -------------
// D# per CDNA5 ISA: 2D tile, tile_dim0=1 (x),
// tile_dim1=rows (y), tensor_dim0_stride=NSPEC elements, data_size=4B.
// Rows beyond tensor_dim1 read as zero (used for the blaze boundary row).
static __device__ inline void tdm_load_column(unsigned lds_byte_off,
                                              const float* gptr,
                                              unsigned tile_rows,
                                              unsigned valid_rows)
{
    unsigned long long ga = (unsigned long long)(size_t)gptr;
    v4u g0;
    g0[0] = 1u;                                   // count=1, user mode
    g0[1] = lds_byte_off;                         // lds_addr (bytes)
    g0[2] = (unsigned)ga;                         // global_addr[31:0]
    g0[3] = ((unsigned)(ga >> 32) & 0x01FFFFFFu)  // global_addr[56:32]
            | (2u << 30);                         // type=2 ("image")
    const unsigned d0 = 1u;                       // tensor_dim0 (x extent)
    const unsigned d1 = valid_rows;               // tensor_dim1 (OOB clip)
    const unsigned t0 = 1u;                       // tile_dim0
    const unsigned t1 = tile_rows;                // tile_dim1
    v8i g1;
    g1[0] = (int)(2u << 16);                      // data_size=2 (4 bytes)
    g1[1] = (int)((d0 & 0xFFFFu) << 16);          // tensor_dim0[15:0] @bit48
    g1[2] = (int)(((d1 & 0xFFFFu) << 16) | ((d0 >> 16) & 0xFFFFu));
    g1[3] = (int)(((t0 & 0xFFFFu) << 16) | ((d1 >> 16) & 0xFFFFu));
    g1[4] = (int)(t1 & 0xFFFFu);                  // tile_dim1; tile_dim2=0
    g1[5] = (int)(unsigned)NSPEC;                 // tensor_dim0_stride lo32
    g1[6] = 0;                                    // stride0 hi / stride1 lo
    g1[7] = 0;                                    // stride1 hi
    v4i z4 = {0, 0, 0, 0};
#if __clang_major__ >= 23
    v8i z8 = {0, 0, 0, 0, 0, 0, 0, 0};
    __builtin_amdgcn_tensor_load_to_lds(g0, g1, z4, z4, z8, 0);
#else
    __builtin_amdgcn_tensor_load_to_lds(g0, g1, z4, z4, 0);
#endif
}

__global__ __launch_bounds__(256)
void star_conv_kernel(const float* __restrict__ star_flux,
                      const float* __restrict__ lab_flux,
                      const float* __restrict__ raw_p,
                      const float* __restrict__ waves_lr,
                      const float* __restrict__ wshift,   // [NXM,NSPEC]
                      const float* __restrict__ weights,
                      const float* __restrict__ lsf,      // [NLSF,NSPEC]
                      const float* __restrict__ data,
                      const float* __restrict__ master,   // wave_hr_master
                      const float* __restrict__ blaze,
                      float* __restrict__ partials)
{
    __shared__ float lsf_pad[LSFPAD];   // zero-padded taps (band mask for free)
    __shared__ float Lbuf[LBUF];        // innerpad[i0-63 .. i0+2113]
    __shared__ float rawL[LBUF];        // TDM-staged raw column (same indexing)
    __shared__ float blzL[CHUNK + 1];   // TDM-staged blaze column [i0 .. i0+2048]
    __shared__ float Mbuf[CHUNK + 1];   // blaze * conv (+1 boundary)
    __shared__ float redA[256];
    __shared__ float redB[256];

    const int tid = threadIdx.x;
    const int s   = blockIdx.y;
    const int i0  = blockIdx.x * CHUNK;

    // ---- wave 0: kick off async TDM column loads (overlap with interp) ----
    if (tid < 32) {
        // raw window: rows [i0-63, i0+2113], clipped to [0, NXM-1]
        int r0   = i0 - 63;
        int r0c  = max(r0, 0);
        int rend = min(i0 + LBUF - 64, NXM - 1);
        unsigned rows = (unsigned)(rend - r0c + 1);
        tdm_load_column((unsigned)(size_t)(const void*)&rawL[0]
                            + (unsigned)(r0c - r0) * 4u,
                        raw_p + (size_t)r0c * NSPEC + s, rows, rows);
        // blaze window: tile 2049 rows; valid rows clipped -> OOB row = 0.0
        unsigned bval = (unsigned)min(CHUNK + 1, NXM - i0);
        tdm_load_column((unsigned)(size_t)(const void*)&blzL[0],
                        blaze + (size_t)i0 * NSPEC + s, CHUNK + 1, bval);
    }

    // ---- stage zero-padded LSF ----
    if (tid < LSFPAD) {
        float v = 0.0f;
        int t = tid - 16;
        if (t >= 0 && t < NLSF) v = lsf[t * NSPEC + s];
        lsf_pad[tid] = v;
    }

    // ---- pass 1: star interp onto master grid (latency-heavy gathers) ----
    for (int u = tid; u < LBUF; u += 256) {
        int j = i0 - 63 + u;
        float val = 1.0f;                 // pad value outside model grid
        if (j >= 0 && j < NXM) {
            float x   = master[j];
            float wsj = wshift[(size_t)j * NSPEC + s];
            // analytic guess (doppler <= ~12 cells), exact searchsorted fixup
            int g = j + (int)floorf((x - wsj) * (1.0f / DWM));
            g = min(max(g, 0), NXM - 2);
            while (g < NXM - 2 && wshift[(size_t)(g + 1) * NSPEC + s] < x) g++;
            while (g > 0 && wshift[(size_t)g * NSPEC + s] >= x) g--;
            float x0 = wshift[(size_t)g * NSPEC + s];
            float x1 = wshift[(size_t)(g + 1) * NSPEC + s];
            float y0 = star_flux[g];
            float y1 = star_flux[g + 1];
            val = y0 + (y1 - y0) / (x1 - x0) * (x - x0);   // star value
        }
        Lbuf[u] = val;
    }

    // ---- TDM completion, then make staged columns visible to all waves ----
    if (tid < 32) __builtin_amdgcn_s_wait_tensorcnt(0);
    __syncthreads();

    // ---- pass 2: inner = star*raw + lab (raw from LDS; own slots only) ----
    for (int u = tid; u < LBUF; u += 256) {
        int j = i0 - 63 + u;
        if (j >= 0 && j < NXM)
            Lbuf[u] = Lbuf[u] * rawL[u] + lab_flux[j];
    }
    __syncthreads();

    const int wave = tid >> 5;
    const int lane = tid & 31;
    const int half = lane >> 4;     // K sub-pair selector (ISA 16x4 A layout)
    const int l16  = lane & 15;     // M for A, N for B
    const int p0   = wave * 256;

    // ---- boundary output conv[i0+CHUNK]: wave-0 shuffle reduction ----------
    if (wave == 0) {
        float p = 0.0f;
        for (int k = lane; k < NLSF; k += 32)          // fixed per-lane order
            p += lsf_pad[16 + k] * Lbuf[CHUNK + k];
        #pragma unroll
        for (int off = 16; off > 0; off >>= 1)         // fixed tree order
            p += __shfl_xor(p, off, 32);
        if (lane == 0)
            Mbuf[CHUNK] = blzL[CHUNK] * p;             // OOB row is 0.0 via TDM
    }

    // ---- conv[i0+p] = sum_k lsf[k]*Lbuf[p+k] as banded-Toeplitz WMMA ------
    // out[16n+m] = sum_{q=0}^{143} A[m][q]*Lbuf[p0+16n+q], A[m][q]=lsf[q-m]
    const int abase = 16 + 2 * half - l16;   // in [1,18] -> lsf_pad in-range
    const int u0    = p0 + 16 * l16 + 2 * half;  // even -> b64-mergeable reads
    v8f acc = {};
    #pragma unroll
    for (int kb = 0; kb < 36; ++kb) {
        v2f a;
        a.x = lsf_pad[abase + 4 * kb];
        a.y = lsf_pad[abase + 4 * kb + 1];
        v2f b;
        b.x = Lbuf[u0 + 4 * kb];
        b.y = Lbuf[u0 + 4 * kb + 1];
        // (neg_a, A, neg_b, B, c_mod, C, reuse_a, reuse_b)
        acc = __builtin_amdgcn_wmma_f32_16x16x4_f32(false, a, false, b,
                                                    (short)0, acc, false, false);
    }

    // D layout: lane's 8 acc regs are 8 consecutive outputs at p0+16*l16+8*half
    {
        int pD = p0 + 16 * l16 + 8 * half;     // multiple of 8 -> 16B aligned
        float4 lo, hi;
        lo.x = blzL[pD + 0] * acc[0];
        lo.y = blzL[pD + 1] * acc[1];
        lo.z = blzL[pD + 2] * acc[2];
        lo.w = blzL[pD + 3] * acc[3];
        hi.x = blzL[pD + 4] * acc[4];
        hi.y = blzL[pD + 5] * acc[5];
        hi.z = blzL[pD + 6] * acc[6];
        hi.w = blzL[pD + 7] * acc[7];
        *(float4*)&Mbuf[pD]     = lo;
        *(float4*)&Mbuf[pD + 4] = hi;
    }
    __syncthreads();

    // ---- detector-grid interp + weighted residuals (exactly-once owned) ----
    float num = 0.0f, den = 0.0f;
    if (tid < 96) {                       // conservative window (~68 needed)
        float w_lo = W0 + (float)i0 * DWM;
        int d0 = (int)floorf((w_lo - 5001.0f - 0.05f) * (1.0f / DLR));
        int d = d0 + tid;
        if (d >= 0 && d < NXD) {
            float wl = waves_lr[(size_t)d * NSPEC + s];
            int ind = (int)floorf((wl - W0) * (1.0f / DWM));
            ind = min(max(ind, 0), NXM - 2);
            while (ind < NXM - 2 && master[ind + 1] < wl) ind++;
            while (ind > 0 && master[ind] >= wl) ind--;
            if (ind >= i0 && ind < i0 + CHUNK) {       // this block owns (d,s)
                float m0 = Mbuf[ind - i0];
                float m1 = Mbuf[ind - i0 + 1];
                float x0 = master[ind];
                float x1 = master[ind + 1];
                float model = m0 + (m1 - m0) / (x1 - x0) * (wl - x0);
                float diff = model - data[(size_t)d * NSPEC + s];
                float w = weights[(size_t)d * NSPEC + s];
                num = diff * diff * w;
                den = w;
            }
        }
    }
    redA[tid] = num;
    redB[tid] = den;
    __syncthreads();
    for (int st = 128; st > 0; st >>= 1) {   // fixed-order tree -> deterministic
        if (tid < st) { redA[tid] += redA[tid + st]; redB[tid] += redB[tid + st]; }
        __syncthreads();
    }
    if (tid == 0) {
        int wg = blockIdx.y * gridDim.x + blockIdx.x;
        partials[2 * wg]     = redA[0];
        partials[2 * wg + 1] = redB[0];
    }
}

__global__ __launch_bounds__(256)
void finalize_kernel(const float* __restrict__ partials, int npairs,
                     float* __restrict__ out)
{
    __shared__ float rA[256], rB[256];
    int tid = threadIdx.x;
    float a = 0.0f, b = 0.0f;
    for (int i = tid; i < npairs; i += 256) {   // fixed per-thread order
        a += partials[2 * i];
        b += partials[2 * i + 1];
    }
    rA[tid] = a; rB[tid] = b;
    __syncthreads();
    for (int st = 128; st > 0; st >>= 1) {
        if (tid < st) { rA[tid] += rA[tid + st]; rB[tid] += rB[tid + st]; }
        __syncthreads();
    }
    if (tid == 0) out[0] = sqrtf(rA[0] / rB[0]);
}

extern "C" void kernel_launch(void* const* d_in, const int* in_sizes, int n_in,
                              void* d_out, int out_size, void* d_ws, size_t ws_size,
                              hipStream_t stream)
{
    const float* star   = (const float*)d_in[0];   // star_flux            [65536]
    const float* lab    = (const float*)d_in[1];   // lab_flux             [65536]
    const float* raw    = (const float*)d_in[2];   // raw_models_partial   [65536,128]
    const float* wlr    = (const float*)d_in[3];   // waves_lr             [2048,128]
    const float* wsh    = (const float*)d_in[4];   // waves_hr_star_shifted[65536,128]
    const float* wts    = (const float*)d_in[5];   // weights              [2048,128]
    const float* lsfp   = (const float*)d_in[6];   // lsf                  [129,128]
    const float* data   = (const float*)d_in[7];   // data                 [2048,128]
    const float* master = (const float*)d_in[8];   // wave_hr_master       [65536]
    const float* blaze  = (const float*)d_in[9];   // blaze                [65536,128]

    float* partials = (float*)d_ws;                // 32 KB used

    dim3 grid(NXM / CHUNK, NSPEC);                 // 32 x 128 workgroups
    star_conv_kernel<<<grid, 256, 0, stream>>>(star, lab, raw, wlr, wsh, wts,
                                               lsfp, data, master, blaze, partials);
    int npairs = (NXM / CHUNK) * NSPEC;
    finalize_kernel<<<1, 256, 0, stream>>>(partials, npairs, (float*)d_out);
}